// DiagonalMatrixModel_7593502179815
// MI455X (gfx1250) — compile-verified
//
#include <hip/hip_runtime.h>
#include <math.h>

// out[i,j] = logsumexp_k( diag(diag)[i,k] + x[k,j] )
//          = m + log( exp(diag[i]+x[i,j]-m) + exp(M_j-m)*S_j - exp(x[i,j]-m) ),
//   M_j = max_k x[k,j],  S_j = sum_k exp(x[k,j]-M_j),  m = max(M_j, diag[i]+x[i,j])
//
// Pass 2 (S_j) runs on the matrix pipe: V_WMMA_F32_16X16X4_F32 with A == ones(16x4),
// so every row of D = column sums of B. Sum is invariant to K-slot assignment.

#define SIZE 512
#define NCOL 512

typedef __attribute__((ext_vector_type(2))) float v2f;
typedef __attribute__((ext_vector_type(8))) float v8f;

// ---------------------------------------------------------------- pass 1: M_j
__global__ void lme_colmax(const float* __restrict__ x, float* __restrict__ M) {
    const int col = blockIdx.x * blockDim.x + threadIdx.x;   // 512 threads total
    float m = -INFINITY;
    #pragma unroll 8
    for (int k = 0; k < SIZE; ++k) {
        m = fmaxf(m, x[k * NCOL + col]);                     // coalesced across lanes
    }
    M[col] = m;
}

// --------------------------------------------- pass 2: S_j via f32 WMMA reduce
// One wave handles 16 columns. A = ones(16x4) so D(i,n) = sum_k B(k,n) for all i.
// Lanes 0-15 supply K-slots {0,1} of the 4-chunk, lanes 16-31 supply {2,3};
// with all-ones A the accumulated sum over K=0..511 is independent of slot order.
__global__ void lme_colsum_wmma(const float* __restrict__ x,
                                const float* __restrict__ M,
                                float* __restrict__ S) {
    const int lane   = threadIdx.x & 31;
    const int waveId = (blockIdx.x * blockDim.x + threadIdx.x) >> 5;  // 0..31
    const int col    = (waveId << 4) + (lane & 15);
    const int koff   = (lane >> 4) << 1;          // 0 or 2

    const float m = M[col];

    v2f a; a.x = 1.0f; a.y = 1.0f;                // all-ones A matrix
    v8f c = {};

    #pragma unroll 4
    for (int kb = 0; kb < SIZE; kb += 4) {
        v2f b;
        b.x = __expf(x[(kb + koff + 0) * NCOL + col] - m);
        b.y = __expf(x[(kb + koff + 1) * NCOL + col] - m);
        // 8 args: (neg_a, A, neg_b, B, c_mod, C, reuse_a, reuse_b)
        c = __builtin_amdgcn_wmma_f32_16x16x4_f32(
                false, a, false, b, (short)0, c, false, false);
    }

    // every row of D holds the column sum; lane L (<16) owns column col in c[0]
    if (lane < 16) S[col] = c[0];
}

// ------------------------------------------------------------ pass 3: epilogue
__global__ void lme_final(const float* __restrict__ x,
                          const float* __restrict__ diag,
                          const float* __restrict__ M,
                          const float* __restrict__ S,
                          float* __restrict__ out) {
    const int idx = blockIdx.x * blockDim.x + threadIdx.x;   // 512*512 threads
    const int i = idx >> 9;          // row
    const int j = idx & (NCOL - 1);  // col

    const float xij = x[idx];
    const float Mj  = M[j];
    const float Sj  = S[j];
    const float a   = diag[i] + xij;           // the k==i (diagonal) term
    const float m   = fmaxf(Mj, a);

    // sum_{k!=i} exp(x[k,j]-m) + exp(a-m)  (>= exp(a-m) > 0)
    const float val = __expf(a - m) + __expf(Mj - m) * Sj - __expf(xij - m);
    out[idx] = m + __logf(val);
}

extern "C" void kernel_launch(void* const* d_in, const int* in_sizes, int n_in,
                              void* d_out, int out_size, void* d_ws, size_t ws_size,
                              hipStream_t stream) {
    const float* x    = (const float*)d_in[0];   // [512, 512] row-major
    const float* diag = (const float*)d_in[1];   // [512]
    float*       out  = (float*)d_out;           // [512, 512]

    float* ws_M = (float*)d_ws;                  // 512 floats
    float* ws_S = ws_M + NCOL;                   // 512 floats

    // pass 1: column max (512 threads)
    lme_colmax<<<2, 256, 0, stream>>>(x, ws_M);

    // pass 2: column exp-sums on the WMMA pipe (32 waves = 8 blocks x 4 waves)
    lme_colsum_wmma<<<8, 128, 0, stream>>>(x, ws_M, ws_S);

    // pass 3: elementwise log-domain combine (fully coalesced)
    lme_final<<<(SIZE * NCOL) / 256, 256, 0, stream>>>(x, diag, ws_M, ws_S, out);
}